// GlobalAttention_79130477461747
// MI455X (gfx1250) — compile-verified
//
#include <hip/hip_runtime.h>

// GlobalAttention for MI455X (gfx1250, wave32).
// B=8, C=256, H=W=48 (HW=2304), E=64.
// - All GEMMs via V_WMMA_F32_16X16X4_F32 (fp32-exact vs reference).
// - All LDS panel staging via GLOBAL_LOAD_ASYNC_TO_LDS_B128/B64 (ASYNCcnt),
//   double-buffered: stage t+1 copies in flight while stage t computes;
//   s_wait_asynccnt <= (#loads of stage t+1) == "stage t resident"
//   (async completions retire in order).
// - Spatial attention is "transpose attention" (softmax per row i, output
//   indexed by j) -> two-pass fusion; 2304^2 score matrix never hits HBM.

#define B_  8
#define C_  256
#define H_  48
#define W_  48
#define HW_ (H_ * W_)   // 2304
#define E_  64

typedef float v2f __attribute__((ext_vector_type(2)));
typedef float v8f __attribute__((ext_vector_type(8)));

__device__ __forceinline__ v8f wmma4(v2f a, v2f b, v8f c) {
  // D(16x16,f32) = A(16x4,f32) * B(4x16,f32) + C
  return __builtin_amdgcn_wmma_f32_16x16x4_f32(
      false, a, false, b, (short)0, c, false, false);
}

// async copies global -> LDS (GVS mode: SGPR64 base + VGPR32 byte offset)
__device__ __forceinline__ void async_cp_b128(unsigned lds_byte,
                                              unsigned gl_byte,
                                              const void* sbase) {
  asm volatile("global_load_async_to_lds_b128 %0, %1, %2"
               :: "v"(lds_byte), "v"(gl_byte), "s"(sbase)
               : "memory");
}
__device__ __forceinline__ void async_cp_b64(unsigned lds_byte,
                                             unsigned gl_byte,
                                             const void* sbase) {
  asm volatile("global_load_async_to_lds_b64 %0, %1, %2"
               :: "v"(lds_byte), "v"(gl_byte), "s"(sbase)
               : "memory");
}
template <int N>
__device__ __forceinline__ void async_wait_le() {
  asm volatile("s_wait_asynccnt %0" :: "i"(N) : "memory");
}
__device__ __forceinline__ unsigned lds_off(const void* p) {
  return (unsigned)(size_t)p;  // flat LDS addr[31:0] == workgroup LDS offset
}

// ---------------------------------------------------------------- GAP ----
__global__ void gap_kernel(const float* __restrict__ feat,
                           float* __restrict__ gap) {
  __shared__ float red[256];
  int bc = blockIdx.x;  // b*C + c
  const float* p = feat + (size_t)bc * HW_;
  float s = 0.f;
  for (int i = threadIdx.x; i < HW_; i += 256) s += p[i];
  red[threadIdx.x] = s;
  __syncthreads();
  for (int off = 128; off > 0; off >>= 1) {
    if ((int)threadIdx.x < off) red[threadIdx.x] += red[threadIdx.x + off];
    __syncthreads();
  }
  if (threadIdx.x == 0) gap[bc] = red[0] * (1.0f / (float)HW_);
}

// --------------------------------------------- conv1d(k=3) + sigmoid ----
__global__ void conv_sig_kernel(const float* __restrict__ gap,
                                const float* __restrict__ wq1, const float* __restrict__ bq1,
                                const float* __restrict__ wk1, const float* __restrict__ bk1,
                                float* __restrict__ qr, float* __restrict__ kr) {
  int b = blockIdx.x, c = threadIdx.x;
  const float* g = gap + b * C_;
  float gm = (c > 0)      ? g[c - 1] : 0.f;
  float g0 = g[c];
  float gp = (c < C_ - 1) ? g[c + 1] : 0.f;
  float q = wq1[0] * gm + wq1[1] * g0 + wq1[2] * gp + bq1[0];
  float k = wk1[0] * gm + wk1[1] * g0 + wk1[2] * gp + bk1[0];
  qr[b * C_ + c] = 1.f / (1.f + __expf(-q));
  kr[b * C_ + c] = 1.f / (1.f + __expf(-k));
}

// --------------------- att_c[b,i,j] = softmax_i( kr[i]*qr[j] ) ----------
__global__ void attc_kernel(const float* __restrict__ qr,
                            const float* __restrict__ kr,
                            float* __restrict__ attc) {
  __shared__ float red[256];
  int j = blockIdx.x, b = blockIdx.y, i = threadIdx.x;
  float t = kr[b * C_ + i] * qr[b * C_ + j];
  red[i] = t;
  __syncthreads();
  for (int off = 128; off > 0; off >>= 1) {
    if (i < off) red[i] = fmaxf(red[i], red[i + off]);
    __syncthreads();
  }
  float mx = red[0];
  __syncthreads();
  float e = __expf(t - mx);
  red[i] = e;
  __syncthreads();
  for (int off = 128; off > 0; off >>= 1) {
    if (i < off) red[i] += red[i + off];
    __syncthreads();
  }
  attc[((size_t)b * C_ + i) * C_ + j] = e / red[0];
}

// ------------------------------------------------ generic WMMA GEMM ----
// Out[b,m,n] = sum_k A[b,m,k] * B[b,k,n]   (row-major, ldA=K, ldB=ldOut=N)
// mode 0: plain   mode 1: +bias[m]
// mode 2: fg = fgate*ggate; Out = fg*(acc+bias[m]) + fg   (gated combine)
__global__ __launch_bounds__(256) void wmma_gemm(
    const float* __restrict__ A, const float* __restrict__ Bm,
    const float* __restrict__ bias,
    const float* __restrict__ fgate, const float* __restrict__ ggate,
    float* __restrict__ Out,
    int M, int N, int K,
    long sA, long sB, long sO, long sG, int mode) {
  __shared__ float As[2][64][20];  // 80B rows: B128-aligned, banks spread
  __shared__ float Bs[2][16][36];  // 144B rows: 8B-aligned for B64
  int tid = threadIdx.x;
  int lane = tid & 31, wave = tid >> 5;
  int wm = wave >> 1, wn = wave & 1;  // 4x2 waves -> 64x32 block tile
  int half = lane >> 4;
  int koff = half * 2;                // A/B frag K offset per ISA layout
  int col  = lane & 15;
  int n0 = blockIdx.x * 32, m0 = blockIdx.y * 64;
  long batch = blockIdx.z;
  const float* Ab = A + batch * sA;
  const float* Bb = Bm + batch * sB;

  // per-thread async-copy coordinates: 1x B128 (A) + 1x B64 (B) per stage
  int ar = tid >> 2, aq = tid & 3;    // A: 64 rows x 4 x 16B chunks
  int br = tid >> 4, bq = tid & 15;   // B: 16 rows x 16 x 8B chunks
  unsigned aLds = lds_off(&As[0][0][0]) + (unsigned)(ar * 80 + aq * 16);
  unsigned bLds = lds_off(&Bs[0][0][0]) + (unsigned)(br * 144 + bq * 8);
  unsigned aG0  = (unsigned)(((m0 + ar) * K + aq * 4) * 4);
  unsigned bG0  = (unsigned)((br * N + n0 + bq * 2) * 4);
  const unsigned ABUF = 64 * 20 * 4, BBUF = 16 * 36 * 4;

  auto issue = [&](int buf, int k0) {
    async_cp_b128(aLds + (unsigned)buf * ABUF, aG0 + (unsigned)(k0 * 4), Ab);
    async_cp_b64(bLds + (unsigned)buf * BBUF, bG0 + (unsigned)(k0 * N * 4), Bb);
  };

  v8f acc = {0, 0, 0, 0, 0, 0, 0, 0};
  issue(0, 0);  // prologue

  for (int k0 = 0; k0 < K; k0 += 16) {
    int buf = (k0 >> 4) & 1;
    if (k0 + 16 < K) {
      issue(buf ^ 1, k0 + 16);  // next panel in flight during compute
      async_wait_le<2>();       // in-order retire => current panel resident
    } else {
      async_wait_le<0>();
    }
    if (k0 + 32 < K)  // warm GL2 two stages ahead (global_prefetch_b8)
      __builtin_prefetch(Ab + (size_t)(m0 + ar) * K + k0 + 32 + aq * 4, 0, 0);
    __syncthreads();
#pragma unroll
    for (int kk = 0; kk < 16; kk += 4) {
      v2f a, bf;
      a.x  = As[buf][wm * 16 + col][kk + koff];
      a.y  = As[buf][wm * 16 + col][kk + koff + 1];
      bf.x = Bs[buf][kk + koff][wn * 16 + col];
      bf.y = Bs[buf][kk + koff + 1][wn * 16 + col];
      acc = wmma4(a, bf, acc);
    }
    __syncthreads();  // all waves done reading buf -> reusable next stage
  }
#pragma unroll
  for (int r = 0; r < 8; r++) {  // D layout: VGPR r -> rows r / r+8
    int m = m0 + wm * 16 + r + half * 8;
    int n = n0 + wn * 16 + col;
    float v = acc[r];
    if (mode >= 1) v += bias[m];
    long oidx = batch * sO + (long)m * N + n;
    if (mode == 2) {
      long gidx = batch * sG + (long)m * N + n;
      float fg = fgate[gidx] * ggate[gidx];
      v = fg * v + fg;
    }
    Out[oidx] = v;
  }
}

// ------------- spatial attention pass 1: row stats m_i, 1/l_i ----------
// S[i,j] = sum_e kp[e,i]*qp[e,j]; per row i: m = max_j, l = sum_j exp(S-m)
__global__ __launch_bounds__(128) void attn_stats_kernel(
    const float* __restrict__ kp, const float* __restrict__ qp,
    float* __restrict__ ms, float* __restrict__ linv) {
  __shared__ float Qs[2][64 * 16];  // 64B rows, B128-aligned
  int tid = threadIdx.x, lane = tid & 31, wave = tid >> 5;
  int half = lane >> 4, koff = half * 2, col = lane & 15;
  int b = blockIdx.y;
  int i0 = blockIdx.x * 64 + wave * 16;
  const float* Kb = kp + (size_t)b * E_ * HW_;
  const float* Qb = qp + (size_t)b * E_ * HW_;

  v2f aK[16];  // A = K^T (16 i x 64 e), preloaded once
#pragma unroll
  for (int ec = 0; ec < 16; ec++) {
    int e = ec * 4 + koff;
    aK[ec].x = Kb[(size_t)e * HW_ + i0 + col];
    aK[ec].y = Kb[(size_t)(e + 1) * HW_ + i0 + col];
  }
  // async-copy coords: 256 chunks of 16B per stage, 2 per thread
  unsigned QsB = lds_off(&Qs[0][0]);
  int c0 = tid, c1 = tid + 128;
  unsigned q0L = QsB + (unsigned)(c0 * 16), q1L = QsB + (unsigned)(c1 * 16);
  unsigned q0G = (unsigned)((((c0 >> 2) * HW_) + (c0 & 3) * 4) * 4);
  unsigned q1G = (unsigned)((((c1 >> 2) * HW_) + (c1 & 3) * 4) * 4);
  const unsigned QBUF = 64 * 16 * 4;

  auto issue = [&](int buf, int j0) {
    unsigned g = (unsigned)(j0 * 4);
    async_cp_b128(q0L + (unsigned)buf * QBUF, q0G + g, Qb);
    async_cp_b128(q1L + (unsigned)buf * QBUF, q1G + g, Qb);
  };

  float mrow[8], lrow[8];
#pragma unroll
  for (int r = 0; r < 8; r++) { mrow[r] = -1e30f; lrow[r] = 0.f; }

  issue(0, 0);
  for (int j0 = 0; j0 < HW_; j0 += 16) {
    int buf = (j0 >> 4) & 1;
    if (j0 + 16 < HW_) {
      issue(buf ^ 1, j0 + 16);
      async_wait_le<2>();
    } else {
      async_wait_le<0>();
    }
    __syncthreads();
    v8f s = {0, 0, 0, 0, 0, 0, 0, 0};
#pragma unroll
    for (int ec = 0; ec < 16; ec++) {
      v2f bq2;
      bq2.x = Qs[buf][(ec * 4 + koff) * 16 + col];
      bq2.y = Qs[buf][(ec * 4 + koff + 1) * 16 + col];
      s = wmma4(aK[ec], bq2, s);
    }
#pragma unroll
    for (int r = 0; r < 8; r++) {
      float v = s[r];
      float mx = v;
#pragma unroll
      for (int d = 8; d >= 1; d >>= 1) mx = fmaxf(mx, __shfl_xor(mx, d, 16));
      float mnew = fmaxf(mrow[r], mx);
      float alpha = __expf(mrow[r] - mnew);
      float p = __expf(v - mnew);
      float sum = p;
#pragma unroll
      for (int d = 8; d >= 1; d >>= 1) sum += __shfl_xor(sum, d, 16);
      lrow[r] = lrow[r] * alpha + sum;
      mrow[r] = mnew;
    }
    __syncthreads();
  }
  if (col == 0) {
#pragma unroll
    for (int r = 0; r < 8; r++) {
      int i = i0 + r + half * 8;
      ms[(size_t)b * HW_ + i]   = mrow[r];
      linv[(size_t)b * HW_ + i] = 1.f / lrow[r];
    }
  }
}

// -- pass 2: gse[e,j] = sum_i vp[e,i] * exp(S[i,j]-m_i)/l_i -------------
__global__ __launch_bounds__(128) void attn_apply_kernel(
    const float* __restrict__ kp, const float* __restrict__ qp,
    const float* __restrict__ vp,
    const float* __restrict__ ms, const float* __restrict__ linv,
    float* __restrict__ gse) {
  __shared__ float Qs[64 * 68];        // 272B rows; 64e x 64j block panel
  __shared__ float Ks[2][64 * 16];     // 64B rows; per i-tile, double-buffered
  __shared__ float Vs[2][64 * 20];     // 80B rows
  __shared__ float Ps[4][16 * 17];     // per-wave P tile bounce buffer
  int tid = threadIdx.x, lane = tid & 31, wave = tid >> 5;
  int half = lane >> 4, koff = half * 2, col = lane & 15;
  int b = blockIdx.y;
  int jb0 = blockIdx.x * 64;
  const float* Kb = kp + (size_t)b * E_ * HW_;
  const float* Qb = qp + (size_t)b * E_ * HW_;
  const float* Vb = vp + (size_t)b * E_ * HW_;
  const float* mb = ms + (size_t)b * HW_;
  const float* lb = linv + (size_t)b * HW_;
  float* sP = &Ps[wave][0];

  // one-time Q panel (64x64): 1024 16B chunks, 8 per thread
  unsigned QsB = lds_off(Qs), KsB = lds_off(&Ks[0][0]), VsB = lds_off(&Vs[0][0]);
#pragma unroll
  for (int u = 0; u < 8; u++) {
    int c = tid + u * 128;
    int row = c >> 4, q = c & 15;
    async_cp_b128(QsB + (unsigned)(row * 272 + q * 16),
                  (unsigned)((row * HW_ + jb0 + q * 4) * 4), Qb);
  }
  // per i-tile K/V: 256 chunks each per stage, 2+2 per thread
  int c0 = tid, c1 = tid + 128;
  unsigned k0L = KsB + (unsigned)(c0 * 16), k1L = KsB + (unsigned)(c1 * 16);
  unsigned v0L = VsB + (unsigned)((c0 >> 2) * 80 + (c0 & 3) * 16);
  unsigned v1L = VsB + (unsigned)((c1 >> 2) * 80 + (c1 & 3) * 16);
  unsigned g0  = (unsigned)((((c0 >> 2) * HW_) + (c0 & 3) * 4) * 4);
  unsigned g1  = (unsigned)((((c1 >> 2) * HW_) + (c1 & 3) * 4) * 4);
  const unsigned KBUF = 64 * 16 * 4, VBUF = 64 * 20 * 4;

  auto issue = [&](int buf, int it0) {
    unsigned g = (unsigned)(it0 * 4);
    async_cp_b128(k0L + (unsigned)buf * KBUF, g0 + g, Kb);
    async_cp_b128(k1L + (unsigned)buf * KBUF, g1 + g, Kb);
    async_cp_b128(v0L + (unsigned)buf * VBUF, g0 + g, Vb);
    async_cp_b128(v1L + (unsigned)buf * VBUF, g1 + g, Vb);
  };

  v8f zero = {0, 0, 0, 0, 0, 0, 0, 0};
  v8f o[4] = {zero, zero, zero, zero};  // O tile: 64e x 16j per wave

  issue(0, 0);  // Q panel (8 older ops) drains under the first wait<=4
  for (int it0 = 0; it0 < HW_; it0 += 16) {
    int buf = (it0 >> 4) & 1;
    if (it0 + 16 < HW_) {
      issue(buf ^ 1, it0 + 16);
      async_wait_le<4>();
    } else {
      async_wait_le<0>();
    }
    __syncthreads();
    // S tile (16i x 16j): A = K^T[i,e] from Ks, B = Q[e,j] from Qs
    v8f s = zero;
#pragma unroll
    for (int ec = 0; ec < 16; ec++) {
      v2f a, bq2;
      a.x   = Ks[buf][(ec * 4 + koff) * 16 + col];
      a.y   = Ks[buf][(ec * 4 + koff + 1) * 16 + col];
      bq2.x = Qs[(ec * 4 + koff) * 68 + wave * 16 + col];
      bq2.y = Qs[(ec * 4 + koff + 1) * 68 + wave * 16 + col];
      s = wmma4(a, bq2, s);
    }
    // P = exp(S - m_i) * (1/l_i); D-layout -> LDS (i x j, pitch 17)
#pragma unroll
    for (int r = 0; r < 8; r++) {
      int i = it0 + r + half * 8;
      float p = __expf(s[r] - mb[i]) * lb[i];
      sP[(r + half * 8) * 17 + col] = p;
    }
    // O += V(e x i) @ P(i x j): A from Vs, B from sP
#pragma unroll
    for (int ec = 0; ec < 4; ec++) {
#pragma unroll
      for (int ii = 0; ii < 16; ii += 4) {
        v2f aV, bP;
        int e = ec * 16 + col;
        aV.x = Vs[buf][e * 20 + ii + koff];
        aV.y = Vs[buf][e * 20 + ii + koff + 1];
        bP.x = sP[(ii + koff) * 17 + col];
        bP.y = sP[(ii + koff + 1) * 17 + col];
        o[ec] = wmma4(aV, bP, o[ec]);
      }
    }
    __syncthreads();  // buf free for stage it0+32
  }
#pragma unroll
  for (int ec = 0; ec < 4; ec++) {
#pragma unroll
    for (int r = 0; r < 8; r++) {
      int e = ec * 16 + r + half * 8;
      int j = jb0 + wave * 16 + col;
      gse[(size_t)b * E_ * HW_ + (size_t)e * HW_ + j] = o[ec][r];
    }
  }
}

// ----------------------------------------------------------- launch ----
extern "C" void kernel_launch(void* const* d_in, const int* in_sizes, int n_in,
                              void* d_out, int out_size, void* d_ws, size_t ws_size,
                              hipStream_t stream) {
  (void)in_sizes; (void)n_in; (void)out_size; (void)ws_size;
  const float* feature = (const float*)d_in[0];
  const float* wq1 = (const float*)d_in[1];
  const float* bq1 = (const float*)d_in[2];
  const float* wk1 = (const float*)d_in[3];
  const float* bk1 = (const float*)d_in[4];
  const float* wq2 = (const float*)d_in[5];
  const float* bq2 = (const float*)d_in[6];
  const float* wk2 = (const float*)d_in[7];
  const float* bk2 = (const float*)d_in[8];
  const float* wv2 = (const float*)d_in[9];
  const float* bv2 = (const float*)d_in[10];
  const float* watt = (const float*)d_in[11];
  const float* batt = (const float*)d_in[12];
  float* out = (float*)d_out;
  float* ws = (float*)d_ws;

  size_t off = 0;
  auto alloc = [&](size_t n) {
    float* p = ws + off;
    off += (n + 63) & ~(size_t)63;
    return p;
  };
  float* gap  = alloc((size_t)B_ * C_);
  float* qr   = alloc((size_t)B_ * C_);
  float* kr   = alloc((size_t)B_ * C_);
  float* attc = alloc((size_t)B_ * C_ * C_);      //  2 MB
  float* gc   = alloc((size_t)B_ * HW_ * C_);     // 19 MB
  float* qp   = alloc((size_t)B_ * E_ * HW_);     // 4.7 MB
  float* kp   = alloc((size_t)B_ * E_ * HW_);
  float* vp   = alloc((size_t)B_ * E_ * HW_);
  float* gse  = alloc((size_t)B_ * E_ * HW_);
  float* msb  = alloc((size_t)B_ * HW_);
  float* lib  = alloc((size_t)B_ * HW_);          // total ~ 40 MB

  // ---- channel attention ----
  gap_kernel<<<B_ * C_, 256, 0, stream>>>(feature, gap);
  conv_sig_kernel<<<B_, C_, 0, stream>>>(gap, wq1, bq1, wk1, bk1, qr, kr);
  attc_kernel<<<dim3(C_, B_), C_, 0, stream>>>(qr, kr, attc);
  // gc(hw x c) = value(hw x c) @ att_c(c x c); value = raw-reshaped feature
  wmma_gemm<<<dim3(C_ / 32, HW_ / 64, B_), 256, 0, stream>>>(
      feature, attc, nullptr, nullptr, nullptr, gc,
      HW_, C_, C_, (long)C_ * HW_, (long)C_ * C_, (long)HW_ * C_, 0, 0);

  // ---- spatial attention: projections (E x HW) = W(E x C) @ F(C x HW) ----
  wmma_gemm<<<dim3(HW_ / 32, E_ / 64, B_), 256, 0, stream>>>(
      wq2, feature, bq2, nullptr, nullptr, qp,
      E_, HW_, C_, 0, (long)C_ * HW_, (long)E_ * HW_, 0, 1);
  wmma_gemm<<<dim3(HW_ / 32, E_ / 64, B_), 256, 0, stream>>>(
      wk2, feature, bk2, nullptr, nullptr, kp,
      E_, HW_, C_, 0, (long)C_ * HW_, (long)E_ * HW_, 0, 1);
  wmma_gemm<<<dim3(HW_ / 32, E_ / 64, B_), 256, 0, stream>>>(
      wv2, feature, bv2, nullptr, nullptr, vp,
      E_, HW_, C_, 0, (long)C_ * HW_, (long)E_ * HW_, 0, 1);

  // ---- fused transpose-attention (score matrix never hits HBM) ----
  attn_stats_kernel<<<dim3(HW_ / 64, B_), 128, 0, stream>>>(kp, qp, msb, lib);
  attn_apply_kernel<<<dim3(HW_ / 64, B_), 128, 0, stream>>>(kp, qp, vp, msb, lib, gse);

  // ---- out = watt @ gse + batt, fused gate: f*gc*gs + f*gc ----
  wmma_gemm<<<dim3(HW_ / 32, C_ / 64, B_), 256, 0, stream>>>(
      watt, gse, batt, feature, gc, out,
      C_, HW_, E_, 0, (long)E_ * HW_, (long)C_ * HW_, (long)C_ * HW_, 2);
}